// CTRGC_44263932952853
// MI455X (gfx1250) — compile-verified
//
#include <hip/hip_runtime.h>

typedef __attribute__((ext_vector_type(2))) float v2f;
typedef __attribute__((ext_vector_type(8))) float v8f;

// D = A(16x4) * B(4x16) + C, fp32 WMMA (CDNA5)
#define WMMA_F32_4(a, b, c) \
  __builtin_amdgcn_wmma_f32_16x16x4_f32(false, (a), false, (b), (short)0, (c), false, false)

constexpr int N_   = 128;
constexpr int CIN  = 64;
constexpr int COUT = 64;
constexpr int CINT = 16;
constexpr int T_   = 128;
constexpr int V_   = 25;
constexpr int TV   = T_ * V_;   // 3200
constexpr int VV   = V_ * V_;   // 625
constexpr int TB   = 16;        // t-rows per fused block
constexpr int NCOL = TB * V_;   // 400 g-tile columns
constexpr int GSTR = NCOL + 2;  // 402: stride%8==2 -> conflict-free D stores
constexpr int ADST = 632;       // per-wave A_dyn staging stride

// ---------------------------------------------------------------- kernel 1
// x_mean[n,c,v] = mean_t x[n,c,t,v]   (one wave per (n,c) pair)
__global__ __launch_bounds__(256) void k_xmean(const float* __restrict__ x,
                                               float* __restrict__ xmean) {
  const int w = threadIdx.x >> 5, lane = threadIdx.x & 31;
  const int pair = blockIdx.x * 8 + w;  // n*CIN + c
  if (lane < V_) {
    const float* xp = x + (size_t)pair * TV + lane;
    float s = 0.f;
#pragma unroll 4
    for (int t = 0; t < T_; ++t) s += xp[t * V_];
    xmean[(size_t)pair * V_ + lane] = s * (1.f / T_);
  }
}

// ---------------------------------------------------------------- kernel 2
// A_dyn[n,o,u,v] = (A+PA)[u,v] + alpha*(br[o] + sum_i Wr[o,i]*tanh(th[i,u]-ph[i,v]))
__global__ __launch_bounds__(256) void k_adyn(
    const float* __restrict__ xmean, const float* __restrict__ A,
    const float* __restrict__ PA, const float* __restrict__ alpha,
    const float* __restrict__ Wth, const float* __restrict__ bth,
    const float* __restrict__ Wph, const float* __restrict__ bph,
    const float* __restrict__ Wr, const float* __restrict__ br,
    float* __restrict__ adyn) {
  __shared__ float xm_l[CIN * V_];   // 1600
  __shared__ float th_l[CINT * V_];  // 400
  __shared__ float ph_l[CINT * V_];  // 400
  __shared__ float apa_l[VV];        // 625
  __shared__ float rel_l[CINT * VV]; // 10000
  const int tid = threadIdx.x;
  const int n = blockIdx.x;

  for (int i = tid; i < CIN * V_; i += 256) xm_l[i] = xmean[(size_t)n * CIN * V_ + i];
  for (int i = tid; i < VV; i += 256) apa_l[i] = A[i] + PA[i];
  __syncthreads();

  for (int i = tid; i < CINT * V_; i += 256) {
    const int ii = i / V_, v = i % V_;
    float st = bth[ii], sp = bph[ii];
    for (int c = 0; c < CIN; ++c) {
      const float xv = xm_l[c * V_ + v];
      st += Wth[ii * CIN + c] * xv;
      sp += Wph[ii * CIN + c] * xv;
    }
    th_l[i] = st;
    ph_l[i] = sp;
  }
  __syncthreads();

  for (int i = tid; i < CINT * VV; i += 256) {
    const int ii = i / VV, uv = i % VV, u = uv / V_, v = uv % V_;
    rel_l[i] = tanhf(th_l[ii * V_ + u] - ph_l[ii * V_ + v]);
  }
  __syncthreads();

  const float al = alpha[0];
  float* ao = adyn + (size_t)n * COUT * VV;
  for (int i = tid; i < COUT * VV; i += 256) {
    const int o = i / VV, uv = i % VV;
    float s = br[o];
#pragma unroll
    for (int ii = 0; ii < CINT; ++ii) s += Wr[o * CINT + ii] * rel_l[ii * VV + uv];
    ao[i] = apa_l[uv] + al * s;
  }
}

// ---------------------------------------------------------------- kernel 3
// Phase 1: g_tile[64 x 400] = Wg @ x_tile + bg   (WMMA, g kept in LDS)
// Phase 2: out[t,u] = sum_v g[c,t,v] * A_dyn[c,u,v]  per channel (WMMA)
__global__ __launch_bounds__(256) void k_fused(const float* __restrict__ x,
                                               const float* __restrict__ Wg,
                                               const float* __restrict__ bg,
                                               const float* __restrict__ adyn,
                                               float* __restrict__ out) {
  __shared__ float g_lds[COUT * GSTR];  // 64*402 fp32  (~100.5 KB)
  __shared__ float wg_lds[COUT * 65];   // padded Wg    (~16.6 KB)
  __shared__ float bg_lds[COUT];
  __shared__ float ad_lds[8 * ADST];    // per-wave A_dyn slice (~19.8 KB)

  const int tid  = threadIdx.x;
  const int lane = tid & 31;
  const int w    = tid >> 5;
  const int half = lane >> 4;  // 0: lanes 0-15, 1: lanes 16-31
  const int lm   = lane & 15;
  const int n    = blockIdx.x >> 3;  // T_/TB = 8 t-blocks per batch
  const int tb   = blockIdx.x & 7;
  const int t0   = tb * TB;

  // stage Wg (row stride 65 -> conflict-free A-frag reads) and bg
  for (int i = tid; i < COUT * CIN; i += 256)
    wg_lds[(i >> 6) * 65 + (i & 63)] = Wg[i];
  if (tid < COUT) bg_lds[tid] = bg[tid];
  __syncthreads();

  // ----- phase 1: M=64 (4 tiles), N=400 (25 tiles), K=64 (16 wmma steps)
  const int mi  = w & 3;   // this wave's c_out 16-row tile
  const int ct0 = w >> 2;  // starting column tile (0/1), stride 2

  v2f afr[16];  // Wg fragments, reused across all column tiles
  {
    const int m = mi * 16 + lm;
#pragma unroll
    for (int kk = 0; kk < 16; ++kk) {
      const int k0 = kk * 4 + half * 2;
      afr[kk].x = wg_lds[m * 65 + k0];
      afr[kk].y = wg_lds[m * 65 + k0 + 1];
    }
  }
  float bfr[8];
#pragma unroll
  for (int r = 0; r < 8; ++r) bfr[r] = bg_lds[mi * 16 + half * 8 + r];

  const float* xn = x + (size_t)n * CIN * TV + (size_t)t0 * V_;
  for (int ct = ct0; ct < NCOL / 16; ct += 2) {
    const int col = ct * 16 + lm;
    const float* xp = xn + (size_t)(half * 2) * TV + col;
    v8f acc;
#pragma unroll
    for (int r = 0; r < 8; ++r) acc[r] = bfr[r];  // bias-initialized C
#pragma unroll
    for (int kk = 0; kk < 16; ++kk) {
      v2f b;
      b.x = xp[(size_t)(kk * 4) * TV];      // coalesced 64B per half-wave
      b.y = xp[(size_t)(kk * 4 + 1) * TV];
      acc = WMMA_F32_4(afr[kk], b, acc);
    }
    const int mb = mi * 16 + half * 8;
#pragma unroll
    for (int r = 0; r < 8; ++r) g_lds[(mb + r) * GSTR + col] = acc[r];
  }
  __syncthreads();

  // ----- phase 2: per channel GEMM  M=16(t) N=25(u, 2 tiles) K=25(v, 7 steps)
  for (int j = 0; j < 8; ++j) {
    const int c = w + 8 * j;  // each wave owns 8 channels
    const float* ad = adyn + ((size_t)n * COUT + c) * VV;
    float* as = ad_lds + w * ADST;
    for (int i2 = lane; i2 < VV; i2 += 32) as[i2] = ad[i2];
    __syncthreads();  // uniform: 8 iterations for every wave

    v8f acc0 = {};
    v8f acc1 = {};
#pragma unroll
    for (int ks = 0; ks < 7; ++ks) {
      const int v0 = ks * 4 + half * 2;  // K index for this lane's pair
      v2f a;  // A = g[t, v], row t = lm
      a.x = (v0 < V_) ? g_lds[c * GSTR + lm * V_ + v0] : 0.f;
      a.y = (v0 + 1 < V_) ? g_lds[c * GSTR + lm * V_ + v0 + 1] : 0.f;
      v2f b0, b1;  // B[v][u] = A_dyn[u][v]; col u = lm (+16)
      b0.x = (v0 < V_) ? as[lm * V_ + v0] : 0.f;
      b0.y = (v0 + 1 < V_) ? as[lm * V_ + v0 + 1] : 0.f;
      const int u1 = 16 + lm;
      b1.x = (u1 < V_ && v0 < V_) ? as[u1 * V_ + v0] : 0.f;
      b1.y = (u1 < V_ && v0 + 1 < V_) ? as[u1 * V_ + v0 + 1] : 0.f;
      acc0 = WMMA_F32_4(a, b0, acc0);
      acc1 = WMMA_F32_4(a, b1, acc1);
    }
    float* op = out + (((size_t)n * COUT + c) * T_ + t0) * V_;
#pragma unroll
    for (int r = 0; r < 8; ++r) {
      const int tr = half * 8 + r;
      op[tr * V_ + lm] = acc0[r];
      const int u1 = 16 + lm;
      if (u1 < V_) op[tr * V_ + u1] = acc1[r];
    }
  }
}

// ---------------------------------------------------------------- launch
extern "C" void kernel_launch(void* const* d_in, const int* in_sizes, int n_in,
                              void* d_out, int out_size, void* d_ws, size_t ws_size,
                              hipStream_t stream) {
  const float* x     = (const float*)d_in[0];
  const float* A     = (const float*)d_in[1];
  const float* PA    = (const float*)d_in[2];
  const float* alpha = (const float*)d_in[3];
  const float* Wg    = (const float*)d_in[4];
  const float* bg    = (const float*)d_in[5];
  const float* Wth   = (const float*)d_in[6];
  const float* bth   = (const float*)d_in[7];
  const float* Wph   = (const float*)d_in[8];
  const float* bph   = (const float*)d_in[9];
  const float* Wr    = (const float*)d_in[10];
  const float* br    = (const float*)d_in[11];
  float* out = (float*)d_out;

  float* xmean = (float*)d_ws;                   // 128*64*25      =   204,800 f
  float* adyn  = xmean + (size_t)N_ * CIN * V_;  // 128*64*625     = 5,120,000 f

  k_xmean<<<(N_ * CIN) / 8, 256, 0, stream>>>(x, xmean);
  k_adyn<<<N_, 256, 0, stream>>>(xmean, A, PA, alpha, Wth, bth, Wph, bph, Wr, br, adyn);
  k_fused<<<N_ * (T_ / TB), 256, 0, stream>>>(x, Wg, bg, adyn, out);
}